// DetectionGenerator_52252572123572
// MI455X (gfx1250) — compile-verified
//
#include <hip/hip_runtime.h>
#include <hip/hip_bf16.h>

#define BB 8
#define NN 50000
#define CC 91
#define TOPK 1000
#define MAXDET 100
#define NBINS 2048
#define CAND_CAP 2048
#define SCORE_THRC 0.05f
#define IOU_THRC 0.5f
#define FLATC (CC * MAXDET) /* 9100 */

#define NEG_INF (-__builtin_inff())

typedef __attribute__((ext_vector_type(4))) unsigned tdm_v4u;
typedef __attribute__((ext_vector_type(8))) int      tdm_v8i;
typedef __attribute__((ext_vector_type(4))) int      tdm_v4i;

// ---------------------------------------------------------------------------
// CDNA5 helpers
// ---------------------------------------------------------------------------
__device__ __forceinline__ unsigned lds_off_u32(const void* p) {
  // addrspacecast(LDS->flat) puts the group-segment byte offset in the low
  // 32 bits; async/TDM LDS addresses are wave-relative byte offsets.
  return (unsigned)(unsigned long long)p;
}

__device__ __forceinline__ void async_copy_b32(unsigned lds_byte_off,
                                               const void* sbase,
                                               unsigned gbyte_off) {
  // GLOBAL_LOAD_ASYNC_TO_LDS_B32, GVS mode: mem = SGPR base + VGPR offset.
  asm volatile("global_load_async_to_lds_b32 %0, %1, %2"
               :: "v"(lds_byte_off), "v"(gbyte_off), "s"(sbase)
               : "memory");
}

__device__ __forceinline__ void wait_async0() {
  asm volatile("s_wait_asynccnt 0x0" ::: "memory");
}

// TDM: 1-D contiguous tile of `elems` 4-byte elements, global -> LDS.
// D# layout per CDNA5 ISA ch.8 (group0: ctrl/lds/global addr; group1: dims).
__device__ __forceinline__ void tdm_load_1d_b32(unsigned lds_byte_off,
                                                const void* gptr,
                                                unsigned elems) {
  unsigned long long ga = (unsigned long long)gptr;
  tdm_v4u g0;
  g0.x = 1u;                                    // count=1 valid descriptor
  g0.y = lds_byte_off;                          // lds_addr[31:0]
  g0.z = (unsigned)(ga & 0xFFFFFFFFu);          // global_addr[31:0]
  g0.w = (unsigned)((ga >> 32) & 0x01FFFFFFu)   // global_addr[56:32]
         | (2u << 30);                          // type=2 ("image")
  tdm_v8i g1;
  g1.s0 = (int)(2u << 16);                      // wg_mask=0, data_size=2 (4B)
  g1.s1 = (int)((elems & 0xFFFFu) << 16);       // tensor_dim0[15:0]
  g1.s2 = (int)(((elems >> 16) & 0xFFFFu))      // tensor_dim0[31:16]
          | (int)(1u << 16);                    // tensor_dim1 = 1
  g1.s3 = (int)((elems & 0xFFFFu) << 16);       // tile_dim0 = elems
  g1.s4 = 1;                                    // tile_dim1 = 1, tile_dim2 = 0
  g1.s5 = (int)elems;                           // tensor_dim0_stride[31:0]
  g1.s6 = 0;
  g1.s7 = 0;
  tdm_v4i gz = {0, 0, 0, 0};                    // groups 2/3 unused (<=2D)
#if defined(__clang_major__) && (__clang_major__ >= 23)
  tdm_v8i gz8 = {0, 0, 0, 0, 0, 0, 0, 0};
  __builtin_amdgcn_tensor_load_to_lds(g0, g1, gz, gz, gz8, 0);
#else
  __builtin_amdgcn_tensor_load_to_lds(g0, g1, gz, gz, 0);
#endif
}

__device__ __forceinline__ void wait_tensor0() {
  __builtin_amdgcn_s_wait_tensorcnt((short)0);
}

// ---------------------------------------------------------------------------
// Kernel A: exact top-1000 per (batch, class) via 2-level radix refine +
// LDS bitonic sort. One block per (b,c). Strided (364 B) reads are
// L2-resident on MI455X: an 18.2 MB per-batch score slab << 192 MB L2 is
// shared by that batch's 91 class blocks, so only the first touch pays HBM.
// ---------------------------------------------------------------------------
__global__ void __launch_bounds__(1024)
det_topk_per_class(const float* __restrict__ scores,
                   float* __restrict__ tk_s, unsigned* __restrict__ tk_i) {
  const int bc = blockIdx.x;            // b*CC + c
  const int b  = bc / CC, c = bc % CC;
  const int tid = threadIdx.x;
  const float* sp = scores + (size_t)b * NN * CC + c;

  __shared__ unsigned hist[NBINS];
  __shared__ float    ck[CAND_CAP];
  __shared__ unsigned cv[CAND_CAP];
  __shared__ unsigned s_counter, s_coarse, s_fine, s_above;

  // ---- pass 1: coarse histogram of score bits[30:20] (monotonic for s>=0)
  for (int i = tid; i < NBINS; i += 1024) hist[i] = 0;
  __syncthreads();
  for (int n = tid; n < NN; n += 1024) {
    if (n + 4096 < NN) __builtin_prefetch(sp + (size_t)(n + 4096) * CC, 0, 1);
    unsigned u = __float_as_uint(sp[(size_t)n * CC]);
    unsigned bin = (u & 0x80000000u) ? 0u : (u >> 20);
    if (bin > NBINS - 1u) bin = NBINS - 1u;
    atomicAdd(&hist[bin], 1u);
  }
  __syncthreads();
  if (tid == 0) {
    unsigned cum = 0; int t = NBINS - 1;
    for (; t > 0; --t) { if (cum + hist[t] >= TOPK) break; cum += hist[t]; }
    s_coarse = (unsigned)t; s_above = cum;
  }
  __syncthreads();
  const unsigned coarse = s_coarse, above = s_above;

  // ---- pass 2: fine histogram of bits[19:9] within the threshold bin
  for (int i = tid; i < NBINS; i += 1024) hist[i] = 0;
  __syncthreads();
  for (int n = tid; n < NN; n += 1024) {
    unsigned u = __float_as_uint(sp[(size_t)n * CC]);
    unsigned bin = (u & 0x80000000u) ? 0u : (u >> 20);
    if (bin == coarse) atomicAdd(&hist[(u >> 9) & 0x7FFu], 1u);
  }
  __syncthreads();
  if (tid == 0) {
    unsigned cum = above; int t = NBINS - 1;
    for (; t > 0; --t) { if (cum + hist[t] >= TOPK) break; cum += hist[t]; }
    s_fine = (unsigned)t; s_counter = 0;
  }
  __syncthreads();
  const unsigned fine = s_fine;

  // ---- pass 3: collect candidates >= refined threshold
  for (int n = tid; n < NN; n += 1024) {
    float s = sp[(size_t)n * CC];
    unsigned u = __float_as_uint(s);
    unsigned bin = (u & 0x80000000u) ? 0u : (u >> 20);
    bool take = (bin > coarse) ||
                (bin == coarse && ((u >> 9) & 0x7FFu) >= fine);
    if (take) {
      unsigned pos = atomicAdd(&s_counter, 1u);
      if (pos < CAND_CAP) { ck[pos] = s; cv[pos] = (unsigned)n; }
    }
  }
  __syncthreads();
  unsigned cnt = s_counter; if (cnt > CAND_CAP) cnt = CAND_CAP;
  for (int i = tid; i < CAND_CAP; i += 1024)
    if ((unsigned)i >= cnt) { ck[i] = NEG_INF; cv[i] = 0u; }
  __syncthreads();

  // ---- LDS bitonic sort, descending (2048 elements, 1024 threads)
  for (unsigned k = 2; k <= CAND_CAP; k <<= 1) {
    for (unsigned j = k >> 1; j > 0; j >>= 1) {
      for (unsigned i = tid; i < CAND_CAP; i += 1024) {
        unsigned ixj = i ^ j;
        if (ixj > i) {
          bool desc = ((i & k) == 0);
          float a = ck[i], bv = ck[ixj];
          bool sw = desc ? (a < bv) : (a > bv);
          if (sw) {
            unsigned va = cv[i], vb = cv[ixj];
            ck[i] = bv; ck[ixj] = a; cv[i] = vb; cv[ixj] = va;
          }
        }
      }
      __syncthreads();
    }
  }

  float*    os = tk_s + (size_t)bc * TOPK;
  unsigned* oi = tk_i + (size_t)bc * TOPK;
  for (int i = tid; i < TOPK; i += 1024) { os[i] = ck[i]; oi[i] = cv[i]; }
}

// ---------------------------------------------------------------------------
// Kernel B: greedy NMS per (b,c). 256 threads own 4 candidates each in
// registers; boxes cached in LDS. The two contiguous 4 KB input tiles
// (top-k scores + indices) are staged into LDS by the Tensor Data Mover
// (one descriptor each, issued by wave 0 only; TDM is a per-wave DMA op).
// ---------------------------------------------------------------------------
__global__ void __launch_bounds__(256)
det_nms_per_class(const float* __restrict__ tk_s, const unsigned* __restrict__ tk_i,
                  const float* __restrict__ boxes,
                  float* __restrict__ sel_s, float* __restrict__ sel_b) {
  const int bc = blockIdx.x;
  const int b  = bc / CC;
  const int tid = threadIdx.x;
  const int lane = tid & 31, wid = tid >> 5;

  __shared__ float    ls[1024];
  __shared__ unsigned li[1024];
  __shared__ float4   lbox[1024];
  __shared__ float    red_s[8];
  __shared__ int      red_i[8];
  __shared__ float    bb_y1, bb_x1, bb_y2, bb_x2;
  __shared__ int      bb_slot;
  __shared__ float    out_s[MAXDET];
  __shared__ float4   out_b[MAXDET];

  const float*    sbase = tk_s + (size_t)bc * TOPK;
  const unsigned* ibase = tk_i + (size_t)bc * TOPK;

  // TDM staging: wave 0 issues two 1-D tile descriptors, waits TENSORcnt.
  if (wid == 0) {
    tdm_load_1d_b32(lds_off_u32(&ls[0]), sbase, TOPK);
    tdm_load_1d_b32(lds_off_u32(&li[0]), ibase, TOPK);
    wait_tensor0();
  }
  __syncthreads();

  // Register-resident scores; gather candidate boxes into LDS.
  float rs[4]; int rslot[4];
  const float* bx = boxes + (size_t)b * NN * 4;
  #pragma unroll
  for (int k = 0; k < 4; ++k) {
    int slot = tid + k * 256;
    rslot[k] = slot;
    float s = (slot < TOPK) ? ls[slot] : NEG_INF;
    rs[k] = s;
    float4 box = make_float4(0.f, 0.f, 0.f, 0.f);
    if (slot < TOPK && s != NEG_INF) {
      unsigned n = li[slot];
      box = *reinterpret_cast<const float4*>(bx + (size_t)n * 4);
    }
    lbox[slot] = box;
  }
  __syncthreads();

  for (int it = 0; it < MAXDET; ++it) {
    // local argmax over 4 owned slots
    float bs = rs[0]; int bi = rslot[0];
    #pragma unroll
    for (int k = 1; k < 4; ++k) if (rs[k] > bs) { bs = rs[k]; bi = rslot[k]; }
    // wave32 argmax
    #pragma unroll
    for (int off = 16; off > 0; off >>= 1) {
      float osc = __shfl_xor(bs, off, 32);
      int   osl = __shfl_xor(bi, off, 32);
      if (osc > bs) { bs = osc; bi = osl; }
    }
    if (lane == 0) { red_s[wid] = bs; red_i[wid] = bi; }
    __syncthreads();
    if (tid == 0) {
      float m = red_s[0]; int mi = red_i[0];
      for (int w = 1; w < 8; ++w) if (red_s[w] > m) { m = red_s[w]; mi = red_i[w]; }
      float4 bbv = lbox[mi];
      bb_y1 = bbv.x; bb_x1 = bbv.y; bb_y2 = bbv.z; bb_x2 = bbv.w;
      bb_slot = mi;
      bool valid = (m >= SCORE_THRC);
      out_s[it] = valid ? m : -1.0f;
      out_b[it] = valid ? bbv : make_float4(0.f, 0.f, 0.f, 0.f);
    }
    __syncthreads();
    const float y1 = bb_y1, x1 = bb_x1, y2 = bb_y2, x2 = bb_x2;
    const int bslot = bb_slot;
    const float area_a = (y2 - y1) * (x2 - x1);
    #pragma unroll
    for (int k = 0; k < 4; ++k) {
      if (rs[k] == NEG_INF) continue;
      if (rslot[k] == bslot) { rs[k] = NEG_INF; continue; }
      float4 cb = lbox[rslot[k]];
      float iy1 = fmaxf(y1, cb.x), ix1 = fmaxf(x1, cb.y);
      float iy2 = fminf(y2, cb.z), ix2 = fminf(x2, cb.w);
      float inter = fmaxf(iy2 - iy1, 0.f) * fmaxf(ix2 - ix1, 0.f);
      float area_b = (cb.z - cb.x) * (cb.w - cb.y);
      float iou = inter / fmaxf(area_a + area_b - inter, 1e-8f);
      if (iou > IOU_THRC) rs[k] = NEG_INF;
    }
    __syncthreads();
  }

  float* ss = sel_s + (size_t)bc * MAXDET;
  float* sb = sel_b + (size_t)bc * MAXDET * 4;
  for (int i = tid; i < MAXDET; i += 256) {
    ss[i] = out_s[i];
    float4 o = out_b[i];
    sb[i * 4 + 0] = o.x; sb[i * 4 + 1] = o.y;
    sb[i * 4 + 2] = o.z; sb[i * 4 + 3] = o.w;
  }
}

// ---------------------------------------------------------------------------
// Kernel C: per-batch final top-100 over 9100 survivors (iterative argmax,
// yields sorted order). The contiguous per-batch score slab is staged into
// LDS via GLOBAL_LOAD_ASYNC_TO_LDS (ASYNCcnt path), then consumed from LDS.
// ---------------------------------------------------------------------------
__global__ void __launch_bounds__(1024)
det_final_topk(const float* __restrict__ sel_s, const float* __restrict__ sel_b,
               float* __restrict__ out) {
  const int b   = blockIdx.x;
  const int tid = threadIdx.x;
  const int lane = tid & 31, wid = tid >> 5;

  __shared__ float lsc[FLATC];          // 9100 floats = 36.4 KB
  __shared__ float red_s[32];
  __shared__ int   red_i[32];
  __shared__ int   s_best;
  __shared__ int   s_valid;

  const float* sp = sel_s + (size_t)b * FLATC;

  // Async-stage the per-batch score slab into LDS.
  #pragma unroll
  for (int k = 0; k < 9; ++k) {
    int slot = tid + k * 1024;
    if (slot < FLATC)
      async_copy_b32(lds_off_u32(&lsc[slot]), sp, (unsigned)(slot * 4));
  }
  wait_async0();
  if (tid == 0) s_valid = 0;
  __syncthreads();

  float rs[9]; int rslot[9];
  #pragma unroll
  for (int k = 0; k < 9; ++k) {
    int slot = tid + k * 1024;
    rslot[k] = slot;
    rs[k] = (slot < FLATC) ? lsc[slot] : NEG_INF;
  }
  __syncthreads();

  float* fin_b = out;                               // 8*100*4 = 3200
  float* fin_s = out + (size_t)BB * MAXDET * 4;     // +800
  float* fin_c = fin_s + (size_t)BB * MAXDET;       // +800
  float* vcnt  = fin_c + (size_t)BB * MAXDET;       // +8

  for (int it = 0; it < MAXDET; ++it) {
    float bs = rs[0]; int bi = rslot[0];
    #pragma unroll
    for (int k = 1; k < 9; ++k) if (rs[k] > bs) { bs = rs[k]; bi = rslot[k]; }
    #pragma unroll
    for (int off = 16; off > 0; off >>= 1) {
      float osc = __shfl_xor(bs, off, 32);
      int   osl = __shfl_xor(bi, off, 32);
      if (osc > bs) { bs = osc; bi = osl; }
    }
    if (lane == 0) { red_s[wid] = bs; red_i[wid] = bi; }
    __syncthreads();
    if (tid == 0) {
      float m = red_s[0]; int mi = red_i[0];
      for (int w = 1; w < 32; ++w) if (red_s[w] > m) { m = red_s[w]; mi = red_i[w]; }
      s_best = mi;
      if (m > -1.0f) s_valid = s_valid + 1;
      int cls = mi / MAXDET, det = mi % MAXDET;
      fin_s[b * MAXDET + it] = m;
      fin_c[b * MAXDET + it] = (float)cls;
      const float* bp = sel_b + ((size_t)(b * CC + cls) * MAXDET + det) * 4;
      float4 bbv = *reinterpret_cast<const float4*>(bp);
      float* op = fin_b + (size_t)b * MAXDET * 4 + (size_t)it * 4;
      op[0] = bbv.x; op[1] = bbv.y; op[2] = bbv.z; op[3] = bbv.w;
    }
    __syncthreads();
    const int best = s_best;
    #pragma unroll
    for (int k = 0; k < 9; ++k) if (rslot[k] == best) rs[k] = NEG_INF;
    __syncthreads();
  }
  if (tid == 0) vcnt[b] = (float)s_valid;
}

// ---------------------------------------------------------------------------
extern "C" void kernel_launch(void* const* d_in, const int* in_sizes, int n_in,
                              void* d_out, int out_size, void* d_ws, size_t ws_size,
                              hipStream_t stream) {
  (void)in_sizes; (void)n_in; (void)out_size; (void)ws_size;
  const float* boxes  = (const float*)d_in[0];   // (B, N, 1, 4)
  const float* scores = (const float*)d_in[1];   // (B, N, C)

  char* ws = (char*)d_ws;
  size_t off = 0;
  float*    tk_s  = (float*)(ws + off);    off += (size_t)BB * CC * TOPK * 4;
  unsigned* tk_i  = (unsigned*)(ws + off); off += (size_t)BB * CC * TOPK * 4;
  float*    sel_s = (float*)(ws + off);    off += (size_t)BB * CC * MAXDET * 4;
  float*    sel_b = (float*)(ws + off);    off += (size_t)BB * CC * MAXDET * 16;
  // total ws use ~7.3 MB

  det_topk_per_class<<<BB * CC, 1024, 0, stream>>>(scores, tk_s, tk_i);
  det_nms_per_class<<<BB * CC, 256, 0, stream>>>(tk_s, tk_i, boxes, sel_s, sel_b);
  det_final_topk<<<BB, 1024, 0, stream>>>(sel_s, sel_b, (float*)d_out);
}